// HTDGBuilder_2276332667285
// MI455X (gfx1250) — compile-verified
//
#include <hip/hip_runtime.h>
#include <hip/hip_bf16.h>

typedef __attribute__((ext_vector_type(2))) float v2f;
typedef __attribute__((ext_vector_type(8))) float v8f;

#define NSEG   8
#define HDIM   256
#define SEG    24      // 3N
#define EPER   90
#define LSTR   260     // LDS row stride in floats (16B-aligned, breaks bank stride)
#define SPB    2       // samples per block: B-operand (weights) shared by 4 WMMAs/step

// ---------------------------------------------------------------------------
// Kernel A: fused projection (fp32 WMMA 16x16x4) + l2norm + pair dots +
// cross-modal edge-attr assembly. One workgroup (8 waves) per SPB samples.
// Wave w computes output columns [16w,16w+16) of Q (node rows 0..15) and of
// K (node rows 8..23) -- exactly the rows the reference ever uses.
// Weight (B-operand) loads are shared across the SPB samples' WMMAs.
// ---------------------------------------------------------------------------
__global__ __launch_bounds__(256)
void htdg_qk_kernel(const float* __restrict__ zt, const float* __restrict__ za,
                    const float* __restrict__ zf,
                    const float* __restrict__ Wq, const float* __restrict__ bq,
                    const float* __restrict__ Wk, const float* __restrict__ bk,
                    const float* __restrict__ emb,
                    float* __restrict__ out, long O2, long Bn)
{
    __shared__ float sN[SPB][SEG * LSTR];                     // ~50 KB
    __shared__ float pQ2[SPB][16][8], pK2[SPB][16][8];        // per-wave partials
    __shared__ float pQKd[SPB][16][8], pQK8[SPB][8][8];

    const long b0 = (long)blockIdx.x * SPB;
    const int  t  = threadIdx.x;

    // Stage SPB samples' node rows (24 x 256 f32 each) into LDS, float4 loads.
    for (int smp = 0; smp < SPB; ++smp) {
        const long b = b0 + smp;
        if (b >= Bn) break;
        for (int i = t; i < SEG * (HDIM / 4); i += 256) {
            int s  = i >> 6;          // row 0..23
            int c4 = i & 63;          // float4 column
            const float* srow = (s < 8)  ? zt + ((b * 8 + s)       * (long)HDIM)
                              : (s < 16) ? za + ((b * 8 + (s - 8)) * (long)HDIM)
                                         : zf + ((b * 8 + (s - 16))* (long)HDIM);
            float4 v = *(const float4*)(srow + c4 * 4);
            *(float4*)(&sN[smp][s * LSTR + c4 * 4]) = v;
        }
    }
    __syncthreads();

    const int lane  = t & 31;
    const int wave  = t >> 5;          // 0..7 -> 16-col output tile
    const int n     = lane & 15;       // M for A operand, N for B/C/D
    const int khalf = lane >> 4;       // 0: K=0,1   1: K=2,3
    const int col0  = wave * 16;

    v8f accq[SPB], acck[SPB];
    #pragma unroll
    for (int smp = 0; smp < SPB; ++smp) {
        accq[smp] = (v8f){0.f,0.f,0.f,0.f,0.f,0.f,0.f,0.f};
        acck[smp] = (v8f){0.f,0.f,0.f,0.f,0.f,0.f,0.f,0.f};
    }

    const float* wqrow = Wq + (long)(col0 + n) * HDIM;   // B operand: row n of W tile
    const float* wkrow = Wk + (long)(col0 + n) * HDIM;

    #pragma unroll 2
    for (int kk = 0; kk < HDIM / 4; ++kk) {
        const int co = 4 * kk + 2 * khalf;               // K-pair for this lane half
        v2f bqv = *(const v2f*)(wqrow + co);             // shared by SPB samples
        v2f bkv = *(const v2f*)(wkrow + co);
        #pragma unroll
        for (int smp = 0; smp < SPB; ++smp) {
            v2f aq = *(const v2f*)(&sN[smp][n * LSTR + co]);
            v2f ak = *(const v2f*)(&sN[smp][(8 + n) * LSTR + co]);
            accq[smp] = __builtin_amdgcn_wmma_f32_16x16x4_f32(
                false, aq, false, bqv, (short)0, accq[smp], false, false);
            acck[smp] = __builtin_amdgcn_wmma_f32_16x16x4_f32(
                false, ak, false, bkv, (short)0, acck[smp], false, false);
        }
    }

    const float bqc = bq[col0 + n];
    const float bkc = bk[col0 + n];

    // Per accumulator register r: lanes 0-15 hold row r, lanes 16-31 row r+8.
    #pragma unroll
    for (int smp = 0; smp < SPB; ++smp) {
        #pragma unroll
        for (int r = 0; r < 8; ++r) {
            float qv = accq[smp][r] + bqc;
            float kv = acck[smp][r] + bkc;
            const int R = r + 8 * khalf;
            float qq = qv * qv, k2 = kv * kv, qk = qv * kv;
            float ksw = __shfl_xor(kv, 16, 32);   // bring K row r+8 to lanes 0-15
            float qk8 = qv * ksw;                 // Q[r]*K[r+8] (valid in khalf==0)
            #pragma unroll
            for (int d = 1; d < 16; d <<= 1) {    // reduce within each 16-lane half
                qq  += __shfl_xor(qq,  d, 32);
                k2  += __shfl_xor(k2,  d, 32);
                qk  += __shfl_xor(qk,  d, 32);
                qk8 += __shfl_xor(qk8, d, 32);
            }
            if (n == 0) {                          // lanes 0 and 16 publish partials
                pQ2[smp][R][wave]  = qq;
                pK2[smp][R][wave]  = k2;
                pQKd[smp][R][wave] = qk;
                if (khalf == 0) pQK8[smp][r][wave] = qk8;
            }
        }
    }
    __syncthreads();

    // Threads 0..SPB*24-1: deterministic cross-wave sum, disc, cross-edge attrs.
    if (t < SPB * 24) {
        const int  smp = t / 24;
        const int  p   = t - smp * 24;
        const long b   = b0 + smp;
        if (b < Bn) {
            int qr, kr, off8;
            if (p < 8)       { qr = p;     kr = p;     off8 = 0; }
            else if (p < 16) { qr = p - 8; kr = p;     off8 = 1; }
            else             { qr = p - 8; kr = p - 8; off8 = 0; }

            float sq2 = 0.f, sk2 = 0.f, num = 0.f;
            #pragma unroll
            for (int w = 0; w < 8; ++w) {
                sq2 += pQ2[smp][qr][w];
                sk2 += pK2[smp][kr][w];
                num += off8 ? pQK8[smp][qr][w] : pQKd[smp][qr][w];
            }
            const float qn   = fmaxf(sqrtf(sq2), 1e-12f);
            const float kn   = fmaxf(sqrtf(sk2), 1e-12f);
            const float dot  = num / (qn * kn);
            const float disc = 1.0f / (1.0f + __expf(dot));   // 1 - sigmoid(dot)
            const int   et   = (disc > 0.4f) ? 4 : 3;

            float attr[16];
            #pragma unroll
            for (int j = 0; j < 8; ++j) attr[j] = emb[et * 8 + j];
            attr[8]  = disc;  attr[9]  = 0.f;  attr[10] = 0.f;
            attr[11] = (float)et * 0.25f;
            attr[12] = 0.f;   attr[13] = 0.f;  attr[14] = 0.f;  attr[15] = 0.f;

            float* dst = out + O2 + ((b * EPER + 42 + (long)p * 2) * 16L);
            #pragma unroll
            for (int j = 0; j < 16; ++j) { dst[j] = attr[j]; dst[16 + j] = attr[j]; }
        }
    }
}

// ---------------------------------------------------------------------------
// Kernel B: node_feats concat-copy (float4) + batch_vec
// ---------------------------------------------------------------------------
__global__ __launch_bounds__(256)
void htdg_nodes_kernel(const float* __restrict__ zt, const float* __restrict__ za,
                       const float* __restrict__ zf,
                       float* __restrict__ out, long O3, long total4)
{
    long i = (long)blockIdx.x * 256 + threadIdx.x;
    if (i >= total4) return;
    long b  = i / (SEG * 64);
    int rem = (int)(i - b * (SEG * 64));
    int s   = rem >> 6;
    int c4  = rem & 63;
    const float* srow = (s < 8)  ? zt + ((b * 8 + s)        * (long)HDIM)
                      : (s < 16) ? za + ((b * 8 + (s - 8))  * (long)HDIM)
                                 : zf + ((b * 8 + (s - 16)) * (long)HDIM);
    float4 v = *(const float4*)(srow + c4 * 4);
    *(float4*)(out + i * 4) = v;
    if (c4 == 0) out[O3 + b * SEG + s] = (float)b;
}

// ---------------------------------------------------------------------------
// Kernel C: edge_index (both rows) + temporal edge attrs (static per modality)
// ---------------------------------------------------------------------------
__global__ __launch_bounds__(256)
void htdg_edges_kernel(const float* __restrict__ emb, float* __restrict__ out,
                       long O1, long O2, long Etot)
{
    long e = (long)blockIdx.x * 256 + threadIdx.x;
    if (e >= Etot) return;
    long b = e / EPER;
    int le = (int)(e - b * EPER);
    int src, dst;
    if (le < 42) {                        // temporal: m-major, i, both dirs
        int m = le / 14, r = le % 14, i = r >> 1, d = r & 1;
        int u = m * 8 + i, v = u + 1;
        src = d ? v : u;  dst = d ? u : v;
        float* a = out + O2 + e * 16;
        #pragma unroll
        for (int j = 0; j < 8; ++j) a[j] = emb[m * 8 + j];
        a[8]  = 0.f;   a[9]  = 0.125f;  a[10] = 1.f;
        a[11] = (float)m * 0.25f;
        a[12] = 0.f;   a[13] = 0.f;     a[14] = 0.f;  a[15] = 0.f;
    } else {                              // cross-modal: attrs come from kernel A
        int ce = le - 42, p = ce >> 1, d = ce & 1, pp = p >> 3, i = p & 7;
        int oa = (pp == 2) ? 8 : 0;
        int ob = (pp == 0) ? 8 : 16;
        int av = oa + i, bv = ob + i;
        src = d ? bv : av;  dst = d ? av : bv;
    }
    long base = b * SEG;
    out[O1 + e]        = (float)(base + src);
    out[O1 + Etot + e] = (float)(base + dst);
}

extern "C" void kernel_launch(void* const* d_in, const int* in_sizes, int n_in,
                              void* d_out, int out_size, void* d_ws, size_t ws_size,
                              hipStream_t stream) {
    const float* zt  = (const float*)d_in[0];
    const float* za  = (const float*)d_in[1];
    const float* zf  = (const float*)d_in[2];
    const float* Wq  = (const float*)d_in[3];
    const float* bq  = (const float*)d_in[4];
    const float* Wk  = (const float*)d_in[5];
    const float* bk  = (const float*)d_in[6];
    const float* emb = (const float*)d_in[7];
    float* out = (float*)d_out;

    const long Bn   = (long)in_sizes[0] / (NSEG * HDIM);   // 8192
    const long NF   = Bn * SEG * HDIM;                     // node_feats elems
    const long Etot = Bn * EPER;                           // total edges
    const long O1   = NF;                                  // edge_index
    const long O2   = NF + 2 * Etot;                       // edge_attr
    const long O3   = O2 + Etot * 16;                      // batch_vec

    const long total4 = NF / 4;
    htdg_nodes_kernel<<<(unsigned)((total4 + 255) / 256), 256, 0, stream>>>(
        zt, za, zf, out, O3, total4);
    htdg_edges_kernel<<<(unsigned)((Etot + 255) / 256), 256, 0, stream>>>(
        emb, out, O1, O2, Etot);
    htdg_qk_kernel<<<(unsigned)((Bn + SPB - 1) / SPB), 256, 0, stream>>>(
        zt, za, zf, Wq, bq, Wk, bk, emb, out, O2, Bn);
}